// CausalBoW_4767413699338
// MI455X (gfx1250) — compile-verified
//
#include <hip/hip_runtime.h>

// ---------------------------------------------------------------------------
// Causal bag-of-words = causal cumulative mean along T.
//   y[b,t,c] = (1/(t+1)) * sum_{s<=t} x[b,s,c]
//
// Two-pass hierarchical scan, memory-bound. L2 is 192 MB and x is only 64 MiB,
// so kernel 1's read pass leaves x L2-resident and kernel 2's re-read hits L2:
// HBM traffic ~= 64 MiB in + 64 MiB out -> ~5.5 us at 23.3 TB/s. Output stores
// are non-temporal (TH_STORE_NT) so y doesn't evict x from L2 between passes.
//
// Intra-tile prefix runs on the CDNA5 matrix pipe: D = L(16x16 lower-tri ones)
// * X + carry, as 4 chained V_WMMA_F32_16X16X4_F32 (K-chunks of 4).
//
// WMMA f32 16x16x4 lane layouts (CDNA5 ISA 7.12.2):
//   A (16x4):  M = lane&15,  K = vgpr + 2*(lane>>4)    -> v2f per lane
//   B (4x16):  N = lane&15,  K = vgpr + 2*(lane>>4)    -> v2f per lane
//   C/D(16x16):N = lane&15,  M = vgpr + 8*(lane>>4)    -> v8f per lane
//
// Row scaling: v_rcp_f32 + 1 Newton step (3 VALU) instead of IEEE divide.
// ---------------------------------------------------------------------------

typedef __attribute__((ext_vector_type(2))) float v2f;
typedef __attribute__((ext_vector_type(8))) float v8f;

#define B_DIM   8
#define T_DIM   4096
#define C_DIM   512
#define CHUNK   128
#define NCHUNKS (T_DIM / CHUNK)   // 32

__device__ __forceinline__ float fast_rcp(float d) {
    float r = __builtin_amdgcn_rcpf(d);
    float e = __builtin_fmaf(-d, r, 1.0f);
    return __builtin_fmaf(r, e, r);
}

// Kernel 1: per-(b,chunk) column sums -> partial[b, chunk, c]
// Also serves to pull x into L2 (regular temporal hint) ahead of kernel 2.
__global__ void cbow_partial(const float* __restrict__ x, float* __restrict__ partial) {
    const int b     = blockIdx.x;
    const int chunk = blockIdx.y;
    const int c     = threadIdx.x;                       // 512 threads, coalesced
    const float* p  = x + ((size_t)b * T_DIM + (size_t)chunk * CHUNK) * C_DIM + c;
    float s = 0.f;
#pragma unroll 8
    for (int t = 0; t < CHUNK; ++t) s += p[(size_t)t * C_DIM];
    partial[((size_t)b * NCHUNKS + chunk) * C_DIM + c] = s;
}

// Kernel 2: WMMA-based tiled prefix scan within each chunk.
// grid (B, NCHUNKS, C/128), block 256 = 8 waves; each wave owns 16 columns.
__global__ void cbow_scan(const float* __restrict__ x,
                          const float* __restrict__ partial,
                          float* __restrict__ y) {
    const int b     = blockIdx.x;
    const int chunk = blockIdx.y;
    const int cg    = blockIdx.z;

    const int lane = threadIdx.x & 31;                   // wave32
    const int wave = threadIdx.x >> 5;
    const int n    = lane & 15;                          // column / M index
    const int hi   = lane >> 4;                          // half-wave select

    const int col = cg * 128 + wave * 16 + n;

    // Inter-chunk carry: sum of all previous chunks' column sums.
    float off = 0.f;
    for (int k = 0; k < chunk; ++k)
        off += partial[((size_t)b * NCHUNKS + k) * C_DIM + col];

    // A = lower-triangular ones, as 4 K-slices of 4 (constants per lane).
    const int m = n;                                     // A's row index
    v2f aK[4];
#pragma unroll
    for (int j = 0; j < 4; ++j) {
        aK[j][0] = ((4 * j + 2 * hi + 0) <= m) ? 1.f : 0.f;
        aK[j][1] = ((4 * j + 2 * hi + 1) <= m) ? 1.f : 0.f;
    }

    // Initial accumulator: C[m][n] = carry[n] for every row m.
    v8f acc;
#pragma unroll
    for (int i = 0; i < 8; ++i) acc[i] = off;

    const float* xb = x + ((size_t)b * T_DIM + (size_t)chunk * CHUNK) * C_DIM + col;
    float*       yb = y + ((size_t)b * T_DIM + (size_t)chunk * CHUNK) * C_DIM + col;

    for (int tt = 0; tt < CHUNK / 16; ++tt) {
        const int t0 = tt * 16;

        // Load B slices: B_j[k][n] = x[t0 + 4j + k][col]  (L2 hits from pass 1)
        v2f bK[4];
#pragma unroll
        for (int j = 0; j < 4; ++j) {
            const int r = t0 + 4 * j + 2 * hi;
            bK[j][0] = xb[(size_t)r * C_DIM];
            bK[j][1] = xb[(size_t)(r + 1) * C_DIM];
        }

        // D = L * X + carry : 4 chained f32 WMMAs (K = 4 each).
#pragma unroll
        for (int j = 0; j < 4; ++j)
            acc = __builtin_amdgcn_wmma_f32_16x16x4_f32(
                false, aK[j], false, bK[j], (short)0, acc, false, false);

        // Scale row t by 1/(t+1); store non-temporally (y is write-once,
        // never re-read -> keep it from evicting x in L2).
        const int tbase = chunk * CHUNK + t0;
#pragma unroll
        for (int r = 0; r < 8; ++r) {
            const int trow  = r + 8 * hi;
            const float inv = fast_rcp((float)(tbase + trow + 1));
            __builtin_nontemporal_store(acc[r] * inv,
                                        yb + (size_t)(t0 + trow) * C_DIM);
        }

        // New carry = row 15 of D (vgpr 7 on lanes 16..31); broadcast per
        // column into all rows of the next tile's accumulator.
        const float carry = __shfl(acc[7], 16 + n, 32);
#pragma unroll
        for (int i = 0; i < 8; ++i) acc[i] = carry;
    }
}

extern "C" void kernel_launch(void* const* d_in, const int* in_sizes, int n_in,
                              void* d_out, int out_size, void* d_ws, size_t ws_size,
                              hipStream_t stream) {
    const float* x = (const float*)d_in[0];
    float*       y = (float*)d_out;
    float* partial = (float*)d_ws;    // B*NCHUNKS*C floats = 512 KB scratch

    cbow_partial<<<dim3(B_DIM, NCHUNKS), dim3(C_DIM), 0, stream>>>(x, partial);
    cbow_scan<<<dim3(B_DIM, NCHUNKS, C_DIM / 128), dim3(256), 0, stream>>>(x, partial, y);
}